// AdvancedQuantumLSTMCell_35253091566147
// MI455X (gfx1250) — compile-verified
//
#include <hip/hip_runtime.h>
#include <hip/hip_bf16.h>

#define N_QUBITS 8
#define N_LAYERS 2
#define QDIM 256           // 2^8
#define WAVES_PER_BLOCK 16 // 16 batch elements per block -> one WMMA M-tile
#define BLOCK_THREADS (WAVES_PER_BLOCK * 32)

typedef float v2f __attribute__((ext_vector_type(2)));
typedef float v8f __attribute__((ext_vector_type(8)));

// ---------------------------------------------------------------------------
// State layout (per wave, per batch element):
//   flattened amplitude index i in [0,256):  i = r*32 + lane,  r = 0..7
//   i bits [4:0] = lane bits, i bits [7:5] = register index bits.
//   Reference qubit q corresponds to bit (7-q) of i.
// ---------------------------------------------------------------------------

// Gate on a register-index bit (qubits 0..2): both pair members in-lane.
__device__ __forceinline__ void apply_gate_regbit(
    float* sre, float* si, int rb,
    float g00r, float g00i, float g01r, float g01i,
    float g10r, float g10i, float g11r, float g11i) {
  const int m = 1 << rb;
#pragma unroll
  for (int r = 0; r < 8; ++r) {
    if (r & m) continue;
    const int r1 = r | m;
    const float a0r = sre[r],  a0i = si[r];
    const float a1r = sre[r1], a1i = si[r1];
    sre[r]  = g00r * a0r - g00i * a0i + g01r * a1r - g01i * a1i;
    si[r]   = g00r * a0i + g00i * a0r + g01r * a1i + g01i * a1r;
    sre[r1] = g10r * a0r - g10i * a0i + g11r * a1r - g11i * a1i;
    si[r1]  = g10r * a0i + g10i * a0r + g11r * a1i + g11i * a1r;
  }
}

// Gate on a lane-index bit (qubits 3..7): partner amplitude via lane shuffle.
__device__ __forceinline__ void apply_gate_lanebit(
    float* sre, float* si, int lb, int lane,
    float g00r, float g00i, float g01r, float g01i,
    float g10r, float g10i, float g11r, float g11i) {
  const bool hi = (lane >> lb) & 1;
  const float q0r = hi ? g10r : g00r, q0i = hi ? g10i : g00i; // coeff of low amp
  const float q1r = hi ? g11r : g01r, q1i = hi ? g11i : g01i; // coeff of high amp
#pragma unroll
  for (int r = 0; r < 8; ++r) {
    const float ar = sre[r], ai = si[r];
    const float pr = __shfl_xor(ar, 1 << lb, 32);
    const float pi = __shfl_xor(ai, 1 << lb, 32);
    const float xr = hi ? pr : ar, xi = hi ? pi : ai; // low-index amplitude
    const float yr = hi ? ar : pr, yi = hi ? ai : pi; // high-index amplitude
    sre[r] = q0r * xr - q0i * xi + q1r * yr - q1i * yi;
    si[r]  = q0r * xi + q0i * xr + q1r * yi + q1i * yr;
  }
}

// CNOT with both control and target on lane bits: controlled lane swap.
__device__ __forceinline__ void cnot_lanebits(float* sre, float* si,
                                              int pc, int pt, int lane) {
  const bool ctl = (lane >> pc) & 1;
#pragma unroll
  for (int r = 0; r < 8; ++r) {
    const float pr = __shfl_xor(sre[r], 1 << pt, 32);
    const float pi = __shfl_xor(si[r], 1 << pt, 32);
    sre[r] = ctl ? pr : sre[r];
    si[r]  = ctl ? pi : si[r];
  }
}

// Z-expectation sign matrix value: B[k][n] = 1 - 2*bit_{7-n}(k), padded to N=16.
__device__ __forceinline__ float zsign(int k, int n) {
  if (n >= N_QUBITS) return 0.0f;
  return ((k >> (7 - n)) & 1) ? -1.0f : 1.0f;
}

__global__ __launch_bounds__(BLOCK_THREADS)
void qlstm_qsim_kernel(const float* __restrict__ inputs,
                       const float* __restrict__ hidden,
                       float* __restrict__ out,
                       int half_out) {
  __shared__ float probsLDS[WAVES_PER_BLOCK * QDIM]; // 16 KB

  const int lane = threadIdx.x & 31;
  const int w    = threadIdx.x >> 5;
  const int base = blockIdx.x * WAVES_PER_BLOCK;
  const int b    = base + w;

  // ---- per-wave statevector in registers: 8 complex amps per lane ----
  float sre[8], si[8];
#pragma unroll
  for (int r = 0; r < 8; ++r) { sre[r] = 0.0f; si[r] = 0.0f; }
  if (lane == 0) sre[0] = 1.0f; // |0...0>

  // ---- 8 RX(inputs[b,q]) gates ----
#pragma unroll
  for (int q = 0; q < N_QUBITS; ++q) {
    const float half = 0.5f * inputs[(size_t)b * N_QUBITS + q];
    const float c = cosf(half), s = sinf(half);
    // RX = [[c, -i s], [-i s, c]]
    const int p = 7 - q;
    if (p >= 5)
      apply_gate_regbit(sre, si, p - 5, c, 0.f, 0.f, -s, 0.f, -s, c, 0.f);
    else
      apply_gate_lanebit(sre, si, p, lane, c, 0.f, 0.f, -s, 0.f, -s, c, 0.f);
  }

  // ---- 2 layers of Rot gates + CNOT ring ----
#pragma unroll
  for (int l = 0; l < N_LAYERS; ++l) {
#pragma unroll
    for (int q = 0; q < N_QUBITS; ++q) {
      const float* wp = hidden + ((size_t)b * N_LAYERS + l) * (N_QUBITS * 3) + q * 3;
      const float phi = wp[0], th = wp[1], om = wp[2];
      const float c = cosf(0.5f * th), s = sinf(0.5f * th);
      const float a  = 0.5f * (phi + om);
      const float bb = 0.5f * (phi - om);
      const float ca = cosf(a), sa = sinf(a);
      const float cb = cosf(bb), sb = sinf(bb);
      // Rot = [[ep*c, -em*s], [conj(em)*s, conj(ep)*c]]
      const float g00r =  c * ca, g00i = -c * sa;
      const float g01r = -s * cb, g01i = -s * sb;
      const float g10r =  s * cb, g10i = -s * sb;
      const float g11r =  c * ca, g11i =  c * sa;
      const int p = 7 - q;
      if (p >= 5)
        apply_gate_regbit(sre, si, p - 5, g00r, g00i, g01r, g01i, g10r, g10i, g11r, g11i);
      else
        apply_gate_lanebit(sre, si, p, lane, g00r, g00i, g01r, g01i, g10r, g10i, g11r, g11i);
    }

    // CNOT ring: CNOT(q, (q+1)%8); control bit pc=7-q, target bit pt=7-t.
    // q=0: pc=regbit2, pt=regbit1 -> swap regs (4,6),(5,7)
    { float t;
      t = sre[4]; sre[4] = sre[6]; sre[6] = t;  t = si[4]; si[4] = si[6]; si[6] = t;
      t = sre[5]; sre[5] = sre[7]; sre[7] = t;  t = si[5]; si[5] = si[7]; si[7] = t; }
    // q=1: pc=regbit1, pt=regbit0 -> swap regs (2,3),(6,7)
    { float t;
      t = sre[2]; sre[2] = sre[3]; sre[3] = t;  t = si[2]; si[2] = si[3]; si[3] = t;
      t = sre[6]; sre[6] = sre[7]; sre[7] = t;  t = si[6]; si[6] = si[7]; si[7] = t; }
    // q=2: pc=regbit0, pt=lanebit4 -> odd regs: swap across lane^16 (unconditional)
#pragma unroll
    for (int r = 1; r < 8; r += 2) {
      sre[r] = __shfl_xor(sre[r], 16, 32);
      si[r]  = __shfl_xor(si[r], 16, 32);
    }
    // q=3..6: both bits in lane index
    cnot_lanebits(sre, si, 4, 3, lane);
    cnot_lanebits(sre, si, 3, 2, lane);
    cnot_lanebits(sre, si, 2, 1, lane);
    cnot_lanebits(sre, si, 1, 0, lane);
    // q=7: pc=lanebit0, pt=regbit2 -> lanes with bit0=1 swap regs (r, r+4)
    { const bool ctl = lane & 1;
#pragma unroll
      for (int r = 0; r < 4; ++r) {
        const float t0 = sre[r], t1 = si[r];
        sre[r]     = ctl ? sre[r + 4] : sre[r];
        si[r]      = ctl ? si[r + 4]  : si[r];
        sre[r + 4] = ctl ? t0 : sre[r + 4];
        si[r + 4]  = ctl ? t1 : si[r + 4];
      }
    }
  }

  // ---- probabilities to LDS: probsLDS[w][i], i = r*32 + lane ----
#pragma unroll
  for (int r = 0; r < 8; ++r)
    probsLDS[w * QDIM + r * 32 + lane] = sre[r] * sre[r] + si[r] * si[r];

  __syncthreads();

  // ---- wave 0: expvals(16x16) = probs(16x256) x Zsign(256x16pad) via WMMA ----
  // fp32 WMMA 16x16x4, 64 chained K-steps. A layout: lane group g = lane>>4
  // holds K = g*2 + vgpr; row M = lane&15. B layout mirrors with col N = lane&15.
  if (threadIdx.x < 32) {
    const int g = lane >> 4;
    const int n = lane & 15;          // = A row M and B column N for this lane
    const float* pr = &probsLDS[n * QDIM];

    v8f acc = {0.f, 0.f, 0.f, 0.f, 0.f, 0.f, 0.f, 0.f};
#pragma unroll 8
    for (int c = 0; c < QDIM / 4; ++c) {
      const int k0 = c * 4 + g * 2;
      v2f av; av[0] = pr[k0];           av[1] = pr[k0 + 1];
      v2f bv; bv[0] = zsign(k0, n);     bv[1] = zsign(k0 + 1, n);
      acc = __builtin_amdgcn_wmma_f32_16x16x4_f32(
          /*neg_a=*/false, av, /*neg_b=*/false, bv,
          /*c_mod=*/(short)0, acc, /*reuse_a=*/false, /*reuse_b=*/false);
    }

    // D layout: vgpr j, lane group g -> row M = j + 8*g, col N = lane&15.
    if (n < N_QUBITS) {
#pragma unroll
      for (int j = 0; j < 8; ++j) {
        const int row = j + 8 * g;
        const size_t bidx = (size_t)(base + row) * N_QUBITS + n;
        const float e = tanhf(acc[j]);
        out[bidx] = e;                       // tuple element 0
        out[(size_t)half_out + bidx] = e;    // tuple element 1 (identical)
      }
    }
  }
}

extern "C" void kernel_launch(void* const* d_in, const int* in_sizes, int n_in,
                              void* d_out, int out_size, void* d_ws, size_t ws_size,
                              hipStream_t stream) {
  const float* inputs = (const float*)d_in[0]; // (B, 8) fp32
  const float* hidden = (const float*)d_in[1]; // (B, 2, 1, 8, 3) fp32
  float* out = (float*)d_out;                  // 2 * B * 8 fp32 (tuple concat)

  const int B = in_sizes[0] / N_QUBITS;        // 16384
  const int half_out = out_size / 2;           // B * 8

  dim3 grid(B / WAVES_PER_BLOCK);
  dim3 block(BLOCK_THREADS);
  qlstm_qsim_kernel<<<grid, block, 0, stream>>>(inputs, hidden, out, half_out);
}